// LNO_56289841381420
// MI455X (gfx1250) — compile-verified
//
#include <hip/hip_runtime.h>
#include <math.h>

// ---------------- problem constants ----------------
#define Bn   16
#define Hh   41
#define Wn   40
#define Cc   48
#define Mm   12          // M1 == M2 == 12
#define NLn  4
#define HWn  (Hh*Wn)     // 1640
#define BCHW (Bn*Cc*HWn) // 1259520
#define EPSf 1e-5f
#define PIf  3.14159265358979323846f
#define DTXf (2.0f*PIf/(float)(Hh-1))
#define DTYf (15.0f/(float)(Wn-1))

typedef float v2f __attribute__((ext_vector_type(2)));
typedef float v8f __attribute__((ext_vector_type(8)));

// CDNA5 f32 matrix core: D = A(16x4) * B(4x16) + C, full fp32 accuracy.
static __device__ __forceinline__ v8f wmma_f32(v2f a, v2f b, v8f c) {
  return __builtin_amdgcn_wmma_f32_16x16x4_f32(false, a, false, b, (short)0, c,
                                               false, false);
}

// ---------------- DFT matrix tables (layer-independent) ----------------
__global__ void LNO_tables(float* FHr, float* FHi, float* FWr, float* FWi,
                           float* IFHr, float* IFHi, float* IFWr, float* IFWi) {
  int g = blockIdx.x * blockDim.x + threadIdx.x;
  if (g < Hh * Hh) {
    int o = g / Hh, h = g % Hh;
    double ang = 2.0 * 3.14159265358979323846 * (double)(o * h) / (double)Hh;
    FHr[g]  = (float)cos(ang);        FHi[g]  = (float)(-sin(ang));
    IFHr[g] = (float)(cos(ang) / Hh); IFHi[g] = (float)(sin(ang) / Hh);
  }
  if (g < Wn * Wn) {
    int x = g / Wn, w = g % Wn;
    double ang = 2.0 * 3.14159265358979323846 * (double)(x * w) / (double)Wn;
    FWr[g]  = (float)cos(ang);        FWi[g]  = (float)(-sin(ang));
    IFWr[g] = (float)(cos(ang) / Wn); IFWi[g] = (float)(sin(ang) / Wn);
  }
}

// ---------------- lift: h = fc0([x, gx, gy]) -> [B,C,H,W] ----------------
__global__ void LNO_build_h(const float* __restrict__ x,
                            const float* __restrict__ w0,
                            const float* __restrict__ b0, float* __restrict__ h) {
  int g = blockIdx.x * blockDim.x + threadIdx.x;
  if (g >= BCHW) return;
  int pix = g % HWn; int bc = g / HWn; int c = bc % Cc; int b = bc / Cc;
  int zh = pix / Wn, xw = pix % Wn;
  float xv = x[b * HWn + pix];
  float gx = (float)zh / (float)(Hh - 1);
  float gy = (float)xw / (float)(Wn - 1);
  h[g] = b0[c] + w0[c * 3 + 0] * xv + w0[c * 3 + 1] * gx + w0[c * 3 + 2] * gy;
}

// ---------------- instance norm per (b,c) over H*W, biased var ----------------
__global__ void LNO_inorm(const float* __restrict__ in, float* __restrict__ out) {
  __shared__ float s1[256], s2[256];
  int bc = blockIdx.x; int t = threadIdx.x;
  const float* p = in + (size_t)bc * HWn;
  float a = 0.f, b = 0.f;
  for (int e = t; e < HWn; e += 256) { float v = p[e]; a += v; b += v * v; }
  s1[t] = a; s2[t] = b; __syncthreads();
  for (int s = 128; s > 0; s >>= 1) {
    if (t < s) { s1[t] += s1[t + s]; s2[t] += s2[t + s]; }
    __syncthreads();
  }
  float mean = s1[0] / (float)HWn;
  float var  = s2[0] / (float)HWn - mean * mean;
  float rstd = rsqrtf(var + EPSf);
  float* q = out + (size_t)bc * HWn;
  for (int e = t; e < HWn; e += 256) q[e] = (p[e] - mean) * rstd;
}

// ---------------- complex DFT along rows: out[bc,o,w]=sum_h M[o,h]*in[bc,h,w] --
__global__ void LNO_dft_rows(const float* __restrict__ inr,
                             const float* __restrict__ ini,
                             const float* __restrict__ Mr,
                             const float* __restrict__ Mi,
                             float* __restrict__ outr, float* __restrict__ outi) {
  int g = blockIdx.x * blockDim.x + threadIdx.x;
  if (g >= BCHW) return;
  int w = g % Wn; int o = (g / Wn) % Hh; size_t bc = g / HWn;
  const float* br = inr + bc * HWn;
  const float* bi = ini ? ini + bc * HWn : nullptr;
  float ar = 0.f, ai = 0.f;
  for (int h = 0; h < Hh; ++h) {
    float mr = Mr[o * Hh + h], mi = Mi[o * Hh + h];
    float vr = br[h * Wn + w];
    float vi = bi ? bi[h * Wn + w] : 0.f;
    ar += mr * vr - mi * vi;
    ai += mr * vi + mi * vr;
  }
  outr[g] = ar; outi[g] = ai;
}

// ---- complex DFT along cols: out[bc,o,x]=sum_w M[x,w]*in[bc,o,w] ----
__global__ void LNO_dft_cols(const float* __restrict__ inr,
                             const float* __restrict__ ini,
                             const float* __restrict__ Mr,
                             const float* __restrict__ Mi,
                             float* __restrict__ outr, float* __restrict__ outi) {
  int g = blockIdx.x * blockDim.x + threadIdx.x;
  if (g >= BCHW) return;
  int x = g % Wn; size_t row = g / Wn;
  const float* br = inr + row * Wn;
  const float* bi = ini + row * Wn;
  float ar = 0.f, ai = 0.f;
  for (int w = 0; w < Wn; ++w) {
    float mr = Mr[x * Wn + w], mi = Mi[x * Wn + w];
    float vr = br[w], vi = bi[w];
    ar += mr * vr - mi * vi;
    ai += mr * vi + mi * vr;
  }
  outr[g] = ar; outi[g] = ai;
}

// ---- inverse DFT along cols, real output only ----
__global__ void LNO_idft_cols_real(const float* __restrict__ inr,
                                   const float* __restrict__ ini,
                                   const float* __restrict__ Mr,
                                   const float* __restrict__ Mi,
                                   float* __restrict__ outr) {
  int g = blockIdx.x * blockDim.x + threadIdx.x;
  if (g >= BCHW) return;
  int x = g % Wn; size_t row = g / Wn;
  const float* br = inr + row * Wn;
  const float* bi = ini + row * Wn;
  float ar = 0.f;
  for (int w = 0; w < Wn; ++w) {
    ar += Mr[x * Wn + w] * br[w] - Mi[x * Wn + w] * bi[w];
  }
  outr[g] = ar;
}

// ---- per-layer precompute: inv1=1/(lam1-p1), E1=exp(p1*TX)  [i,k,p,o] ----
__global__ void LNO_prep1(const float* __restrict__ p1, float* __restrict__ i1r,
                          float* __restrict__ i1i, float* __restrict__ e1r,
                          float* __restrict__ e1i) {
  int g = blockIdx.x * blockDim.x + threadIdx.x;
  const int TOT = Cc * Cc * Mm * Hh;
  if (g >= TOT) return;
  int o = g % Hh; int ikp = g / Hh;
  float pr = p1[ikp * 2 + 0], pi = p1[ikp * 2 + 1];
  int kk = (o <= Hh / 2) ? o : o - Hh;                 // fftfreq(41)
  float mu = 2.0f * PIf * (float)kk / ((float)Hh * DTXf);
  float a = -pr, b = mu - pi;                          // lam - p = i*mu - p
  float n = a * a + b * b;
  i1r[g] = a / n; i1i[g] = -b / n;
  float tt = (float)o * DTXf;                          // TX grid
  float er = expf(pr * tt);
  e1r[g] = er * cosf(pi * tt); e1i[g] = er * sinf(pi * tt);
}

__global__ void LNO_prep2(const float* __restrict__ p2, float* __restrict__ i2r,
                          float* __restrict__ i2i, float* __restrict__ e2r,
                          float* __restrict__ e2i) {
  int g = blockIdx.x * blockDim.x + threadIdx.x;
  const int TOT = Cc * Cc * Mm * Wn;
  if (g >= TOT) return;
  int x = g % Wn; int ikq = g / Wn;
  float pr = p2[ikq * 2 + 0], pi = p2[ikq * 2 + 1];
  int kk = (x < Wn / 2) ? x : x - Wn;                  // fftfreq(40)
  float mu = 2.0f * PIf * (float)kk / ((float)Wn * DTYf);
  float a = -pr, b = mu - pi;
  float n = a * a + b * b;
  i2r[g] = a / n; i2i[g] = -b / n;
  float tt = (float)x * DTYf;                          // TY grid
  float er = expf(pr * tt);
  e2r[g] = er * cosf(pi * tt); e2i[g] = er * sinf(pi * tt);
}

// ---- S[i,k,o,x] = sum_{p,q} res[p,q]*inv1[p,o]*inv2[q,x] (weights only) ----
__global__ void LNO_S(const float* __restrict__ res, const float* __restrict__ i1r,
                      const float* __restrict__ i1i, const float* __restrict__ i2r,
                      const float* __restrict__ i2i, float* __restrict__ Sr,
                      float* __restrict__ Si) {
  __shared__ float rl[2][Mm * Mm], i2l[2][Mm * Wn], i1l[2][Mm * Hh], Tl[2][Mm * Wn];
  int ik = blockIdx.x; int t = threadIdx.x;
  for (int e = t; e < Mm * Mm; e += 256) {
    rl[0][e] = res[((size_t)ik * Mm * Mm + e) * 2 + 0];
    rl[1][e] = res[((size_t)ik * Mm * Mm + e) * 2 + 1];
  }
  for (int e = t; e < Mm * Wn; e += 256) {
    i2l[0][e] = i2r[(size_t)ik * Mm * Wn + e];
    i2l[1][e] = i2i[(size_t)ik * Mm * Wn + e];
  }
  for (int e = t; e < Mm * Hh; e += 256) {
    i1l[0][e] = i1r[(size_t)ik * Mm * Hh + e];
    i1l[1][e] = i1i[(size_t)ik * Mm * Hh + e];
  }
  __syncthreads();
  for (int e = t; e < Mm * Wn; e += 256) {           // T[p,x] = sum_q res*inv2
    int p = e / Wn, x = e % Wn; float ar = 0.f, ai = 0.f;
    for (int q = 0; q < Mm; ++q) {
      float rr = rl[0][p * Mm + q], ri = rl[1][p * Mm + q];
      float vr = i2l[0][q * Wn + x], vi = i2l[1][q * Wn + x];
      ar += rr * vr - ri * vi; ai += rr * vi + ri * vr;
    }
    Tl[0][e] = ar; Tl[1][e] = ai;
  }
  __syncthreads();
  for (int e = t; e < HWn; e += 256) {               // S[o,x] = sum_p inv1*T
    int o = e / Wn, x = e % Wn; float ar = 0.f, ai = 0.f;
    for (int p = 0; p < Mm; ++p) {
      float wr = i1l[0][p * Hh + o], wi = i1l[1][p * Hh + o];
      float vr = Tl[0][p * Wn + x], vi = Tl[1][p * Wn + x];
      ar += wr * vr - wi * vi; ai += wr * vi + wi * vr;
    }
    Sr[(size_t)ik * HWn + e] = ar; Si[(size_t)ik * HWn + e] = ai;
  }
}

// ---- WMMA: per-pixel complex GEMM res1[b,k] = sum_i alpha[b,i]*S[i,k] ----
// one wave per pixel; M=16 (batch, exact), N=48 (3 tiles), K=48 (12 ksteps)
__global__ void LNO_res1_wmma(const float* __restrict__ alr,
                              const float* __restrict__ ali,
                              const float* __restrict__ Sr,
                              const float* __restrict__ Si,
                              float* __restrict__ r1r, float* __restrict__ r1i) {
  int pix = blockIdx.x;
  int L = threadIdx.x, half = L >> 4, lo = L & 15;
  __builtin_prefetch(alr + pix, 0, 0);
  v8f zero = {0, 0, 0, 0, 0, 0, 0, 0};
  v8f rr[3], ii[3], ri[3], ir[3];
#pragma unroll
  for (int n = 0; n < 3; ++n) { rr[n] = zero; ii[n] = zero; ri[n] = zero; ir[n] = zero; }
  for (int ks = 0; ks < 12; ++ks) {
    int i0 = 4 * ks + 2 * half;
    size_t a0 = ((size_t)(lo * Cc + i0)) * HWn + pix;   // A[m=b][k=i]
    v2f ar = {alr[a0], alr[a0 + HWn]};
    v2f am = {ali[a0], ali[a0 + HWn]};
#pragma unroll
    for (int n = 0; n < 3; ++n) {
      size_t b0 = ((size_t)(i0 * Cc + n * 16 + lo)) * HWn + pix; // B[k=i][n=k]
      v2f br = {Sr[b0], Sr[b0 + (size_t)Cc * HWn]};
      v2f bm = {Si[b0], Si[b0 + (size_t)Cc * HWn]};
      rr[n] = wmma_f32(ar, br, rr[n]); ii[n] = wmma_f32(am, bm, ii[n]);
      ri[n] = wmma_f32(ar, bm, ri[n]); ir[n] = wmma_f32(am, br, ir[n]);
    }
  }
#pragma unroll
  for (int n = 0; n < 3; ++n) {
#pragma unroll
    for (int r = 0; r < 8; ++r) {
      int m = r + 8 * half;              // batch row
      int kc = n * 16 + lo;              // out channel
      size_t d = ((size_t)(m * Cc + kc)) * HWn + pix;
      r1r[d] = rr[n][r] - ii[n][r];
      r1i[d] = ri[n][r] + ir[n][r];
    }
  }
}

// ---- WMMA: r2[b,k,p,q] = sum_i res[i,k,p,q]*sum_o inv1[p,o]*sum_x alpha*inv2 --
// one wave per (k,b); C1[o,q] via WMMA (K=x=40 exact), C2 + res reduce in VALU
__global__ void LNO_r2_wmma(const float* __restrict__ alr,
                            const float* __restrict__ ali,
                            const float* __restrict__ i1r,
                            const float* __restrict__ i1i,
                            const float* __restrict__ i2r,
                            const float* __restrict__ i2i,
                            const float* __restrict__ res,
                            float* __restrict__ r2rg, float* __restrict__ r2ig) {
  __shared__ float i2l[2][Mm * Wn], i1l[2][Mm * Hh], c1[2][Mm * 48];
  int k = blockIdx.x % Cc;
  int b = blockIdx.x / Cc;
  int L = threadIdx.x, half = L >> 4, lo = L & 15;
  v8f zero = {0, 0, 0, 0, 0, 0, 0, 0};
  float accr[5] = {0, 0, 0, 0, 0}, acci[5] = {0, 0, 0, 0, 0};
  int qc = lo < Mm ? lo : Mm - 1;                     // clamp pad cols (unstored)
  for (int i = 0; i < Cc; ++i) {
    int ik = i * Cc + k;
    for (int e = L; e < Mm * Wn; e += 32) {
      i2l[0][e] = i2r[(size_t)ik * Mm * Wn + e];
      i2l[1][e] = i2i[(size_t)ik * Mm * Wn + e];
    }
    for (int e = L; e < Mm * Hh; e += 32) {
      i1l[0][e] = i1r[(size_t)ik * Mm * Hh + e];
      i1l[1][e] = i1i[(size_t)ik * Mm * Hh + e];
    }
    __syncthreads();
    const size_t abase = ((size_t)(b * Cc + i)) * HWn;
    for (int mt = 0; mt < 3; ++mt) {
      v8f wrr = zero, wii = zero, wri = zero, wir = zero;
      int o = mt * 16 + lo; int oc = o > Hh - 1 ? Hh - 1 : o;  // clamp pad rows
      for (int ks = 0; ks < 10; ++ks) {
        int x0 = 4 * ks + 2 * half;
        size_t a0 = abase + (size_t)oc * Wn + x0;     // A[m=o][k=x] (contig x)
        v2f ar = {alr[a0], alr[a0 + 1]};
        v2f am = {ali[a0], ali[a0 + 1]};
        v2f br = {i2l[0][qc * Wn + x0], i2l[0][qc * Wn + x0 + 1]}; // B[k=x][n=q]
        v2f bm = {i2l[1][qc * Wn + x0], i2l[1][qc * Wn + x0 + 1]};
        wrr = wmma_f32(ar, br, wrr); wii = wmma_f32(am, bm, wii);
        wri = wmma_f32(ar, bm, wri); wir = wmma_f32(am, br, wir);
      }
#pragma unroll
      for (int r = 0; r < 8; ++r) {
        int o2 = mt * 16 + r + 8 * half;
        if (lo < Mm) {
          c1[0][lo * 48 + o2] = wrr[r] - wii[r];
          c1[1][lo * 48 + o2] = wri[r] + wir[r];
        }
      }
    }
    __syncthreads();
    for (int j = 0; j < 5; ++j) {                     // C2 + residues reduce
      int e = L + 32 * j;
      if (e < Mm * Mm) {
        int p = e / Mm, q2 = e % Mm;
        float c2r = 0.f, c2i = 0.f;
        for (int o = 0; o < Hh; ++o) {
          float wr = i1l[0][p * Hh + o], wi = i1l[1][p * Hh + o];
          float cr = c1[0][q2 * 48 + o], ci = c1[1][q2 * 48 + o];
          c2r += wr * cr - wi * ci; c2i += wr * ci + wi * cr;
        }
        float rr = res[((size_t)ik * Mm * Mm + e) * 2 + 0];
        float rm = res[((size_t)ik * Mm * Mm + e) * 2 + 1];
        accr[j] += rr * c2r - rm * c2i;
        acci[j] += rr * c2i + rm * c2r;
      }
    }
    __syncthreads();
  }
  for (int j = 0; j < 5; ++j) {
    int e = L + 32 * j;
    if (e < Mm * Mm) {
      r2rg[((size_t)(b * Cc + k)) * Mm * Mm + e] = accr[j];
      r2ig[((size_t)(b * Cc + k)) * Mm * Mm + e] = acci[j];
    }
  }
}

// ---- WMMA: x2[kb,i,z,x] = Re{ sum_{b2,q}(sum_p r2*E1)*E2 }/(H*W), += x1buf ----
// one wave per (kb,i, 16x16 output patch); only real part needed -> 2 chains
__global__ void LNO_x2_wmma(const float* __restrict__ r2r,
                            const float* __restrict__ r2i,
                            const float* __restrict__ e1r,
                            const float* __restrict__ e1i,
                            const float* __restrict__ e2r,
                            const float* __restrict__ e2i,
                            float* __restrict__ x1b) {
  __shared__ float ubr[Mm * 16], ubi[Mm * 16];
  int t9 = blockIdx.x % 9; int ki = blockIdx.x / 9;
  int i = ki % Cc; int kb = ki / Cc;
  int mt = t9 / 3, nt = t9 % 3;
  int L = threadIdx.x, half = L >> 4, lo = L & 15;
  v8f zero = {0, 0, 0, 0, 0, 0, 0, 0};
  v8f accR = zero, accI = zero;
  int x = nt * 16 + lo; int xc = x < Wn ? x : Wn - 1;
  for (int b2 = 0; b2 < Cc; ++b2) {
    for (int e = L; e < Mm * 16; e += 32) {           // U[q,z]=sum_p r2*E1 (VALU)
      int q = e / 16, zl = e % 16, z = mt * 16 + zl;
      float ur = 0.f, ui = 0.f;
      if (z < Hh) {
        for (int p = 0; p < Mm; ++p) {
          size_t ridx = ((size_t)(kb * Cc + b2) * Mm + p) * Mm + q;
          float rr = r2r[ridx], rm = r2i[ridx];
          size_t eidx = ((size_t)(b2 * Cc + i) * Mm + p) * Hh + z;
          float er = e1r[eidx], ei = e1i[eidx];
          ur += rr * er - rm * ei; ui += rr * ei + rm * er;
        }
      }
      ubr[e] = ur; ubi[e] = ui;
    }
    __syncthreads();
#pragma unroll
    for (int ks = 0; ks < 3; ++ks) {                  // K = q = 12
      int q0 = 4 * ks + 2 * half;
      v2f ar = {ubr[q0 * 16 + lo], ubr[(q0 + 1) * 16 + lo]}; // A[m=z][k=q]
      v2f am = {ubi[q0 * 16 + lo], ubi[(q0 + 1) * 16 + lo]};
      size_t b0 = ((size_t)(b2 * Cc + i) * Mm + q0) * Wn + xc; // B[k=q][n=x]
      v2f br = {e2r[b0], e2r[b0 + Wn]};
      v2f bm = {e2i[b0], e2i[b0 + Wn]};
      accR = wmma_f32(ar, br, accR);                  // Re += Ur*E2r
      accI = wmma_f32(am, bm, accI);                  // Re -= Ui*E2i (subtract at end)
    }
    __syncthreads();
  }
  const float sc = 1.0f / (float)(Hh * Wn);
#pragma unroll
  for (int r = 0; r < 8; ++r) {
    int z = mt * 16 + r + 8 * half;
    int xo = nt * 16 + lo;
    if (z < Hh && xo < Wn) {
      size_t d = ((size_t)(kb * Cc + i)) * HWn + (size_t)z * Wn + xo;
      x1b[d] += (accR[r] - accI[r]) * sc;             // exclusive per block
    }
  }
}

// ---- h_new = sin( inorm(x1+x2) + conv1x1(h_old) + bias ) ----
__global__ void LNO_conv_sin(const float* __restrict__ xn,
                             const float* __restrict__ hold,
                             const float* __restrict__ cw,
                             const float* __restrict__ cb,
                             float* __restrict__ hnew) {
  int g = blockIdx.x * blockDim.x + threadIdx.x;
  if (g >= BCHW) return;
  int pix = g % HWn; int bo = g / HWn; int o = bo % Cc; int b = bo / Cc;
  float acc = cb[o] + xn[g];
  const float* hp = hold + (size_t)b * Cc * HWn + pix;
  for (int i2 = 0; i2 < Cc; ++i2) acc += cw[o * Cc + i2] * hp[(size_t)i2 * HWn];
  hnew[g] = sinf(acc);
}

// ---- head: out = fc2( sin( fc1(h) ) ), one block per (b,pixel) ----
__global__ void LNO_head(const float* __restrict__ h, const float* __restrict__ w1,
                         const float* __restrict__ b1, const float* __restrict__ w2,
                         const float* __restrict__ b2, float* __restrict__ out) {
  __shared__ float sh[128];
  int pid = blockIdx.x;                 // b*HWn + pix
  int b = pid / HWn; int pix = pid % HWn;
  int t = threadIdx.x;
  const float* hp = h + (size_t)b * Cc * HWn + pix;
  float acc = b1[t];
  for (int c = 0; c < Cc; ++c) acc += w1[t * Cc + c] * hp[(size_t)c * HWn];
  sh[t] = sinf(acc);
  __syncthreads();
  if (t < 2) {
    float a2 = b2[t];
    for (int o = 0; o < 128; ++o) a2 += w2[t * 128 + o] * sh[o];
    out[(size_t)pid * 2 + t] = a2;
  }
}

// =======================================================================
extern "C" void kernel_launch(void* const* d_in, const int* in_sizes, int n_in,
                              void* d_out, int out_size, void* d_ws, size_t ws_size,
                              hipStream_t stream) {
  (void)in_sizes; (void)n_in; (void)out_size; (void)ws_size;
  const float* x      = (const float*)d_in[0];
  const float* fc0_w  = (const float*)d_in[1];
  const float* fc0_b  = (const float*)d_in[2];
  const float* poles1 = (const float*)d_in[3];
  const float* poles2 = (const float*)d_in[4];
  const float* resid  = (const float*)d_in[5];
  const float* conv_w = (const float*)d_in[6];
  const float* conv_b = (const float*)d_in[7];
  const float* fc1_w  = (const float*)d_in[8];
  const float* fc1_b  = (const float*)d_in[9];
  const float* fc2_w  = (const float*)d_in[10];
  const float* fc2_b  = (const float*)d_in[11];
  float* out = (float*)d_out;

  // ---- workspace layout (floats), ~122 MB total, fits in L2-resident regime
  float* ws = (float*)d_ws;
  size_t off = 0;
  auto alloc = [&](size_t n) { float* p = ws + off; off += (n + 15) & ~(size_t)15; return p; };
  float *FHr = alloc(Hh * Hh), *FHi = alloc(Hh * Hh);
  float *IFHr = alloc(Hh * Hh), *IFHi = alloc(Hh * Hh);
  float *FWr = alloc(Wn * Wn), *FWi = alloc(Wn * Wn);
  float *IFWr = alloc(Wn * Wn), *IFWi = alloc(Wn * Wn);
  float *hA = alloc(BCHW), *hB = alloc(BCHW), *hn = alloc(BCHW);
  float *tr = alloc(BCHW), *ti = alloc(BCHW);
  float *alr = alloc(BCHW), *ali = alloc(BCHW);
  float *r1r = alloc(BCHW), *r1i = alloc(BCHW);
  float *x1b = alloc(BCHW), *xn = alloc(BCHW);
  float *Sr = alloc((size_t)Cc * Cc * HWn), *Si = alloc((size_t)Cc * Cc * HWn);
  float *i1r = alloc(Cc * Cc * Mm * Hh), *i1i = alloc(Cc * Cc * Mm * Hh);
  float *e1r = alloc(Cc * Cc * Mm * Hh), *e1i = alloc(Cc * Cc * Mm * Hh);
  float *i2r = alloc(Cc * Cc * Mm * Wn), *i2i = alloc(Cc * Cc * Mm * Wn);
  float *e2r = alloc(Cc * Cc * Mm * Wn), *e2i = alloc(Cc * Cc * Mm * Wn);
  float *r2r = alloc(Bn * Cc * Mm * Mm), *r2i = alloc(Bn * Cc * Mm * Mm);

  const int GE = (BCHW + 255) / 256;   // 4920 blocks for elementwise passes

  LNO_tables<<<(Hh * Hh + 255) / 256, 256, 0, stream>>>(FHr, FHi, FWr, FWi,
                                                        IFHr, IFHi, IFWr, IFWi);
  LNO_build_h<<<GE, 256, 0, stream>>>(x, fc0_w, fc0_b, hA);

  float* hcur = hA; float* hnext = hB;
  for (int l = 0; l < NLn; ++l) {
    const float* p1 = poles1 + (size_t)l * Cc * Cc * Mm * 2;
    const float* p2 = poles2 + (size_t)l * Cc * Cc * Mm * 2;
    const float* rs = resid + (size_t)l * Cc * Cc * Mm * Mm * 2;

    LNO_inorm<<<Bn * Cc, 256, 0, stream>>>(hcur, hn);
    LNO_dft_rows<<<GE, 256, 0, stream>>>(hn, nullptr, FHr, FHi, tr, ti);
    LNO_dft_cols<<<GE, 256, 0, stream>>>(tr, ti, FWr, FWi, alr, ali);

    LNO_prep1<<<(Cc * Cc * Mm * Hh + 255) / 256, 256, 0, stream>>>(p1, i1r, i1i, e1r, e1i);
    LNO_prep2<<<(Cc * Cc * Mm * Wn + 255) / 256, 256, 0, stream>>>(p2, i2r, i2i, e2r, e2i);
    LNO_S<<<Cc * Cc, 256, 0, stream>>>(rs, i1r, i1i, i2r, i2i, Sr, Si);

    LNO_res1_wmma<<<HWn, 32, 0, stream>>>(alr, ali, Sr, Si, r1r, r1i);

    LNO_dft_rows<<<GE, 256, 0, stream>>>(r1r, r1i, IFHr, IFHi, tr, ti);
    LNO_idft_cols_real<<<GE, 256, 0, stream>>>(tr, ti, IFWr, IFWi, x1b);

    LNO_r2_wmma<<<Bn * Cc, 32, 0, stream>>>(alr, ali, i1r, i1i, i2r, i2i, rs, r2r, r2i);
    LNO_x2_wmma<<<Bn * Cc * 9, 32, 0, stream>>>(r2r, r2i, e1r, e1i, e2r, e2i, x1b);

    LNO_inorm<<<Bn * Cc, 256, 0, stream>>>(x1b, xn);
    LNO_conv_sin<<<GE, 256, 0, stream>>>(xn, hcur, conv_w + (size_t)l * Cc * Cc,
                                         conv_b + (size_t)l * Cc, hnext);
    float* tswp = hcur; hcur = hnext; hnext = tswp;
  }

  LNO_head<<<Bn * HWn, 128, 0, stream>>>(hcur, fc1_w, fc1_b, fc2_w, fc2_b, out);
}